// GCN_86758339379236
// MI455X (gfx1250) — compile-verified
//
#include <hip/hip_runtime.h>
#include <hip/hip_bf16.h>

#define GCN_D 256

typedef float v2f __attribute__((ext_vector_type(2)));
typedef float v8f __attribute__((ext_vector_type(8)));

// ---------------------------------------------------------------------------
// GEMM: out[n, 256] = X[n, 256] @ W[256, 256], full fp32 via
// V_WMMA_F32_16X16X4_F32. One 16x16 output tile per wave32; K stepped by 4.
// A-layout (16x4 f32, 2 VGPRs/lane): lanes 0-15 -> K={k,k+1}, lanes 16-31 ->
// K={k+2,k+3} (contiguous float2 per lane). B-layout (4x16, 2 VGPRs): VGPR
// enumerates K within the half selected by the lane half.
// ---------------------------------------------------------------------------
__global__ void __launch_bounds__(256) gcn_gemm_wmma_f32(
    const float* __restrict__ X, const float* __restrict__ W,
    float* __restrict__ out, int n)
{
  const int lane = threadIdx.x & 31;
  const int wave = threadIdx.x >> 5;
  const int wid  = blockIdx.x * 8 + wave;     // global wave id
  const int rowTile = wid >> 4;               // 256/16 = 16 col tiles per row
  const int colTile = wid & 15;
  const int baseM = rowTile * 16;
  const int baseN = colTile * 16;
  if (baseM >= n) return;

  const int ml = lane & 15;                   // M (for A) / N (for B) index
  const int hi = lane >> 4;                   // lane-half: selects K pair

  int rowA = baseM + ml;
  if (rowA >= n) rowA = n - 1;                // clamp; never stored
  const float* __restrict__ arow = X + (size_t)rowA * GCN_D + hi * 2;
  const float* __restrict__ bcol = W + baseN + ml;

  v8f acc = {};
#pragma unroll
  for (int k = 0; k < GCN_D; k += 4) {
    v2f a = *(const v2f*)(arow + k);          // A[m][k+2*hi .. +1]
    const int kb = k + hi * 2;
    v2f b;
    b.x = bcol[(size_t)(kb + 0) * GCN_D];     // W[kb+0][n]
    b.y = bcol[(size_t)(kb + 1) * GCN_D];     // W[kb+1][n]
    // 8 args: (neg_a, A, neg_b, B, c_mod, C, reuse_a, reuse_b)
    acc = __builtin_amdgcn_wmma_f32_16x16x4_f32(
        false, a, false, b, (short)0, acc, false, false);
  }

  // C/D layout: VGPR r -> row baseM + r + 8*hi, col baseN + (lane & 15)
  float* __restrict__ orow = out + baseN + ml;
#pragma unroll
  for (int r = 0; r < 8; ++r) {
    const int row = baseM + r + 8 * hi;
    if (row < n) orow[(size_t)row * GCN_D] = acc[r];
  }
}

// ---------------------------------------------------------------------------
// Edge scatter: out[dst[e]] += sup[src[e]] * w[e].
// 4 edges per 256-thread block; 64 lanes x float4 = one coalesced 1 KB row.
// sup (102 MB) and out (102 MB) both fit in the 192 MB L2, so gathers are
// L2-resident (avg in-degree 32) and the f32 atomics resolve at L2.
// ---------------------------------------------------------------------------
__global__ void __launch_bounds__(256) gcn_edge_scatter(
    const float* __restrict__ sup, const int* __restrict__ esrc,
    const int* __restrict__ edst, const float* __restrict__ ew,
    float* __restrict__ out, int nEdges)
{
  const int t = threadIdx.x;
  const int e = blockIdx.x * 4 + (t >> 6);
  if (e >= nEdges) return;
  const int c = (t & 63) * 4;

  const int   s = esrc[e];
  const int   d = edst[e];
  const float w = ew[e];

  const float4 v = *(const float4*)(sup + (size_t)s * GCN_D + c);
  float* __restrict__ o = out + (size_t)d * GCN_D + c;
  unsafeAtomicAdd(o + 0, v.x * w);   // global_atomic_add_f32, no-return
  unsafeAtomicAdd(o + 1, v.y * w);
  unsafeAtomicAdd(o + 2, v.z * w);
  unsafeAtomicAdd(o + 3, v.w * w);
}

// ---------------------------------------------------------------------------
// Launch: layer1 (gemm -> zero -> scatter), layer2 (gemm -> zero -> scatter)
// ---------------------------------------------------------------------------
extern "C" void kernel_launch(void* const* d_in, const int* in_sizes, int n_in,
                              void* d_out, int out_size, void* d_ws, size_t ws_size,
                              hipStream_t stream)
{
  const float* X    = (const float*)d_in[0];
  const float* ew   = (const float*)d_in[1];
  const float* W1   = (const float*)d_in[2];
  const float* W2   = (const float*)d_in[3];
  const int*   esrc = (const int*)d_in[4];
  const int*   edst = (const int*)d_in[5];

  const int n  = in_sizes[0] / GCN_D;   // 100000 nodes
  const int nE = in_sizes[1];           // 3.2M edges

  float* sup = (float*)d_ws;            // [n, D] support buffer (reused)
  float* h   = sup + (size_t)n * GCN_D; // [n, D] layer-1 output
  float* out = (float*)d_out;

  const int rowTiles   = (n + 15) / 16;
  const int gemmBlocks = (rowTiles * 16 + 7) / 8;   // 8 waves / block
  const int scatBlocks = (nE + 3) / 4;              // 4 edges / block
  const size_t featBytes = (size_t)n * GCN_D * sizeof(float);

  // ---- layer 1: h = A @ (X @ W1) ----
  gcn_gemm_wmma_f32<<<gemmBlocks, 256, 0, stream>>>(X, W1, sup, n);
  hipMemsetAsync(h, 0, featBytes, stream);
  gcn_edge_scatter<<<scatBlocks, 256, 0, stream>>>(sup, esrc, edst, ew, h, nE);

  // ---- layer 2: out = A @ (h @ W2) ----
  gcn_gemm_wmma_f32<<<gemmBlocks, 256, 0, stream>>>(h, W2, sup, n);
  hipMemsetAsync(out, 0, (size_t)out_size * sizeof(float), stream);
  gcn_edge_scatter<<<scatBlocks, 256, 0, stream>>>(sup, esrc, edst, ew, out, nE);
}